// QMAMLModel_14817637171322
// MI455X (gfx1250) — compile-verified
//
#include <hip/hip_runtime.h>
#include <math.h>

typedef __attribute__((ext_vector_type(2))) float v2f;
typedef __attribute__((ext_vector_type(8))) float v8f;

// ---------- constexpr ring-CNOT permutation tables (matches _ring_perm) ----------
struct Rings { int r1[64]; int r2[64]; };

constexpr Rings makeRings() {
  Rings R{};
  for (int rr = 1; rr <= 2; ++rr) {
    int total[64];
    for (int i = 0; i < 64; ++i) total[i] = i;
    for (int w = 0; w < 6; ++w) {
      int t = (w + rr) % 6;
      int nxt[64];
      for (int i = 0; i < 64; ++i) {
        int cbit = (i >> (5 - w)) & 1;
        nxt[i] = total[i ^ (cbit << (5 - t))];
      }
      for (int i = 0; i < 64; ++i) total[i] = nxt[i];
    }
    for (int i = 0; i < 64; ++i) { if (rr == 1) R.r1[i] = total[i]; else R.r2[i] = total[i]; }
  }
  return R;
}

__constant__ Rings RINGS = makeRings();

// ---------- complex / shuffle helpers ----------
__device__ __forceinline__ float2 cmul(float2 a, float2 b) {
  return make_float2(a.x * b.x - a.y * b.y, a.x * b.y + a.y * b.x);
}
__device__ __forceinline__ float2 cadd(float2 a, float2 b) { return make_float2(a.x + b.x, a.y + b.y); }
__device__ __forceinline__ float2 shflx2(float2 v, int m) {
  return make_float2(__shfl_xor(v.x, m, 32), __shfl_xor(v.y, m, 32));
}
__device__ __forceinline__ float2 shfl2(float2 v, int l) {
  return make_float2(__shfl(v.x, l, 32), __shfl(v.y, l, 32));
}

#define RED16(v) do { v += __shfl_xor(v,1,32); v += __shfl_xor(v,2,32); \
                      v += __shfl_xor(v,4,32); v += __shfl_xor(v,8,32); } while(0)

__global__ __launch_bounds__(256) void qmaml_kernel(
    const float* __restrict__ x,  const float* __restrict__ W1,
    const float* __restrict__ b1, const float* __restrict__ ln_g,
    const float* __restrict__ ln_b, const float* __restrict__ W2,
    const float* __restrict__ b2, const float* __restrict__ th_s,
    const float* __restrict__ th_t, const float* __restrict__ Wc,
    const float* __restrict__ bc, float* __restrict__ out, int B)
{
  __shared__ float  sW1[4096];          // 64x64
  __shared__ float  sW2[384];           // 6x64
  __shared__ float  sB1[64], sG[64], sBeta[64];
  __shared__ float  sB2[8], sWc[32], sBc[8];
  __shared__ float2 sGates[72];         // 18 gates x 4 complex entries
  __shared__ float  sZ[768];            // 8 waves x 16 rows x 6

  const int tid = threadIdx.x;

  // ---------------- stage parameters into LDS ----------------
  for (int i = tid; i < 1024; i += 256) ((float4*)sW1)[i] = ((const float4*)W1)[i];
  for (int i = tid; i < 384; i += 256)  sW2[i] = W2[i];
  if (tid < 64) { sB1[tid] = b1[tid]; sG[tid] = ln_g[tid]; sBeta[tid] = ln_b[tid]; }
  if (tid < 6)  sB2[tid] = b2[tid];
  if (tid < 30) sWc[tid] = Wc[tid];
  if (tid < 5)  sBc[tid] = bc[tid];
  if (tid < 18) {
    int l = tid / 6, w = tid % 6;
    const float* th = (l < 2) ? (th_s + (l * 6 + w) * 3) : (th_t + w * 3);
    float phi = th[0], theta = th[1], omega = th[2];
    float c  = cosf(0.5f * theta), s = sinf(0.5f * theta);
    float aa = 0.5f * (phi + omega), bb = 0.5f * (phi - omega);
    float ca = cosf(aa), sa = sinf(aa);
    float cb = cosf(bb), sb = sinf(bb);
    sGates[tid * 4 + 0] = make_float2( ca * c, -sa * c);   //  em*c
    sGates[tid * 4 + 1] = make_float2(-cb * s, -sb * s);   // -ep*s
    sGates[tid * 4 + 2] = make_float2( cb * s, -sb * s);   //  conj(ep)*s
    sGates[tid * 4 + 3] = make_float2( ca * c,  sa * c);   //  conj(em)*c
  }
  __syncthreads();

  const int lane = tid & 31;
  const int wave = tid >> 5;
  const int half = lane >> 4;    // 0 or 1
  const int lm   = lane & 15;
  const int rowBase = blockIdx.x * 128 + wave * 16;

  // ---------------- GEMM1: h(16x64) = x_tile(16x64) @ W1^T via WMMA f32 16x16x4 ----------------
  v8f acc[4] = {};
  int rowA = rowBase + lm; if (rowA >= B) rowA = B - 1;
  const float* xrow = x + (size_t)rowA * 64 + 2 * half;
  #pragma unroll
  for (int kk = 0; kk < 16; ++kk) {
    v2f a = *(const v2f*)(xrow + 4 * kk);
    #pragma unroll
    for (int t = 0; t < 4; ++t) {
      // B^T frag of W1^T == W1 loaded A-style: W1[16t+lm][4kk + 2*half + {0,1}]
      v2f bf = *(const v2f*)(sW1 + (16 * t + lm) * 64 + 4 * kk + 2 * half);
      acc[t] = __builtin_amdgcn_wmma_f32_16x16x4_f32(false, a, false, bf,
                                                     (short)0, acc[t], false, false);
    }
  }

  // ---------------- bias + ReLU; lane holds rows M=8*half+j, col N=16t+lm ----------------
  float hv[4][8];
  #pragma unroll
  for (int t = 0; t < 4; ++t) {
    float bias = sB1[16 * t + lm];
    #pragma unroll
    for (int j = 0; j < 8; ++j) hv[t][j] = fmaxf(acc[t][j] + bias, 0.0f);
  }

  // ---------------- LayerNorm over H=64 (butterfly across 16 lanes per half) ----------------
  float muj[8], inv[8];
  #pragma unroll
  for (int j = 0; j < 8; ++j) {
    float s  = hv[0][j] + hv[1][j] + hv[2][j] + hv[3][j];
    float q2 = hv[0][j]*hv[0][j] + hv[1][j]*hv[1][j] + hv[2][j]*hv[2][j] + hv[3][j]*hv[3][j];
    RED16(s); RED16(q2);
    float mu  = s * 0.015625f;
    float var = q2 * 0.015625f - mu * mu;
    muj[j] = mu;
    inv[j] = rsqrtf(var + 1e-5f);
  }
  #pragma unroll
  for (int t = 0; t < 4; ++t) {
    float g = sG[16 * t + lm], be = sBeta[16 * t + lm];
    #pragma unroll
    for (int j = 0; j < 8; ++j) hv[t][j] = (hv[t][j] - muj[j]) * inv[j] * g + be;
  }

  // ---------------- z = tanh(h @ W2^T + b2), write per-row z to LDS ----------------
  #pragma unroll
  for (int q = 0; q < 6; ++q) {
    float w0 = sW2[q * 64 + lm],      w1 = sW2[q * 64 + 16 + lm];
    float w2 = sW2[q * 64 + 32 + lm], w3 = sW2[q * 64 + 48 + lm];
    float bq = sB2[q];
    #pragma unroll
    for (int j = 0; j < 8; ++j) {
      float p = hv[0][j]*w0 + hv[1][j]*w1 + hv[2][j]*w2 + hv[3][j]*w3;
      RED16(p);
      if (lm == 0) sZ[wave * 96 + (8 * half + j) * 6 + q] = tanhf(p + bq);
    }
  }
  __syncthreads();

  // ---------------- per-lane constants for readout: ws[k] = sum_q Wc[k,q]*zsign(idx,q) ----------------
  float ws0[5], ws1[5];
  #pragma unroll
  for (int k = 0; k < 5; ++k) {
    float t0 = 0.f, t1 = 0.f;
    #pragma unroll
    for (int q = 0; q < 6; ++q) {
      float w = sWc[k * 6 + q];
      int b = 5 - q;
      t0 += ((lane >> b) & 1) ? -w : w;
      t1 += (((lane + 32) >> b) & 1) ? -w : w;
    }
    ws0[k] = t0; ws1[k] = t1;
  }
  const int rA0 = RINGS.r1[lane], rA1 = RINGS.r1[lane + 32];
  const int rB0 = RINGS.r2[lane], rB1 = RINGS.r2[lane + 32];

  // ---------------- quantum simulation: 2 amplitudes per lane (idx = lane, lane+32) ----------------
  for (int r = 0; r < 16; ++r) {
    int row = rowBase + r;
    float cq[6], sq[6];
    #pragma unroll
    for (int q = 0; q < 6; ++q) {
      float h2 = 1.57079632679489662f * sZ[wave * 96 + r * 6 + q];
      cq[q] = cosf(h2); sq[q] = sinf(h2);
    }
    // product state: amp(i) = prod_q (bit ? sin : cos); idx1 differs only in bit 5 (q=0)
    float pr = 1.f;
    #pragma unroll
    for (int q = 1; q < 6; ++q) pr *= ((lane >> (5 - q)) & 1) ? sq[q] : cq[q];
    float2 a0 = make_float2(cq[0] * pr, 0.f);
    float2 a1 = make_float2(sq[0] * pr, 0.f);

    #pragma unroll
    for (int l = 0; l < 3; ++l) {
      #pragma unroll
      for (int w = 0; w < 6; ++w) {
        int g = l * 6 + w;
        float2 m00 = sGates[g * 4 + 0], m01 = sGates[g * 4 + 1];
        float2 m10 = sGates[g * 4 + 2], m11 = sGates[g * 4 + 3];
        if (w == 0) {                       // bit 5: slot-local pair
          float2 n0 = cadd(cmul(m00, a0), cmul(m01, a1));
          float2 n1 = cadd(cmul(m10, a0), cmul(m11, a1));
          a0 = n0; a1 = n1;
        } else {                            // bits 4..0: shuffle partner
          int b = 5 - w, mask = 1 << b;
          float2 p0 = shflx2(a0, mask), p1 = shflx2(a1, mask);
          int bit = (lane >> b) & 1;
          float2 e0 = cadd(cmul(m00, a0), cmul(m01, p0));
          float2 o0 = cadd(cmul(m10, p0), cmul(m11, a0));
          float2 e1 = cadd(cmul(m00, a1), cmul(m01, p1));
          float2 o1 = cadd(cmul(m10, p1), cmul(m11, a1));
          a0 = bit ? o0 : e0;
          a1 = bit ? o1 : e1;
        }
      }
      // ring permutation: psi_new[i] = psi_old[ring[i]]
      int s0 = (l == 1) ? rB0 : rA0;
      int s1 = (l == 1) ? rB1 : rA1;
      float2 g0a = shfl2(a0, s0 & 31), g0b = shfl2(a1, s0 & 31);
      float2 g1a = shfl2(a0, s1 & 31), g1b = shfl2(a1, s1 & 31);
      a0 = (s0 < 32) ? g0a : g0b;
      a1 = (s1 < 32) ? g1a : g1b;
    }

    // readout: out[k] = sum_i |psi_i|^2 * ws_i[k], reduced over all 32 lanes
    float p0 = a0.x * a0.x + a0.y * a0.y;
    float p1 = a1.x * a1.x + a1.y * a1.y;
    float o[5];
    #pragma unroll
    for (int k = 0; k < 5; ++k) {
      float v = p0 * ws0[k] + p1 * ws1[k];
      v += __shfl_xor(v, 1, 32); v += __shfl_xor(v, 2, 32); v += __shfl_xor(v, 4, 32);
      v += __shfl_xor(v, 8, 32); v += __shfl_xor(v, 16, 32);
      o[k] = v;
    }
    if (lane == 0 && row < B) {
      #pragma unroll
      for (int k = 0; k < 5; ++k) out[(size_t)row * 5 + k] = o[k] + sBc[k];
    }
  }
}

extern "C" void kernel_launch(void* const* d_in, const int* in_sizes, int n_in,
                              void* d_out, int out_size, void* d_ws, size_t ws_size,
                              hipStream_t stream) {
  (void)n_in; (void)out_size; (void)d_ws; (void)ws_size;
  const float* x    = (const float*)d_in[0];
  const float* W1   = (const float*)d_in[1];
  const float* b1   = (const float*)d_in[2];
  const float* ln_g = (const float*)d_in[3];
  const float* ln_b = (const float*)d_in[4];
  const float* W2   = (const float*)d_in[5];
  const float* b2   = (const float*)d_in[6];
  const float* th_s = (const float*)d_in[7];
  const float* th_t = (const float*)d_in[8];
  const float* Wc   = (const float*)d_in[9];
  const float* bc   = (const float*)d_in[10];
  float* out = (float*)d_out;
  int B = in_sizes[0] / 64;
  int grid = (B + 127) / 128;
  qmaml_kernel<<<grid, 256, 0, stream>>>(x, W1, b1, ln_g, ln_b, W2, b2, th_s, th_t, Wc, bc, out, B);
}